// CaponAttentionUnit_79877801771096
// MI455X (gfx1250) — compile-verified
//
#include <hip/hip_runtime.h>
#include <math.h>
#include <stdint.h>

#define BB 4
#define CC 64
#define HH 128
#define WW 128
#define KK 5
#define STR 2
#define PADP 2
#define HOUT 64
#define WOUT 64
#define LL (HOUT*WOUT)
#define MM 25
#define MPAD 32
#define HID 32
#define ALPHA 0.1f
#define DELTA 1e-3f
#define XS 33   // Xc row stride (pad to dodge bank conflicts)
#define RS 65   // R row stride

typedef float v2f __attribute__((ext_vector_type(2)));
typedef float v8f __attribute__((ext_vector_type(8)));

__global__ __launch_bounds__(256)
void capon_tile_kernel(const float* __restrict__ U,
                       const float* __restrict__ w1,
                       const float* __restrict__ b1,
                       const float* __restrict__ w2,
                       const float* __restrict__ b2,
                       float* __restrict__ y_out,
                       float* __restrict__ v_out)
{
    __shared__ float Xc[CC * XS];     // patch, K padded to 32 with zeros
    __shared__ float Rm[CC * RS];     // covariance -> Cholesky (lower), R_s kept in upper
    __shared__ float mu[CC], uc[CC], ssq[CC], dsave[CC];
    __shared__ float av[CC], wt[CC], wv[CC], tv[CC];
    __shared__ float hv[HID];
    __shared__ float scal[4];

    const int tid   = threadIdx.x;
    const int bidx  = blockIdx.x;
    const int batch = bidx >> 12;         // / 4096
    const int tile  = bidx & 4095;
    const int ty    = tile >> 6;
    const int tx    = tile & 63;
    const int h0    = ty * STR - PADP;
    const int w0    = tx * STR - PADP;

    // ---------------- zero staging (provides zero-padding for OOB and K=25..31) ----
    for (int e = tid; e < CC * XS; e += 256) Xc[e] = 0.f;
    __syncthreads();

    // ---------------- async global->LDS gather of the 5x5x64 patch ----------------
    // GVS mode: uniform SGPR 64-bit base + per-lane 32-bit byte offset; destination
    // LDS byte offset in a VGPR. Tracked on ASYNCcnt. OOB lanes are EXEC-masked and
    // keep the pre-zeroed values.
    {
        const unsigned long long ubase =
            (unsigned long long)(const void*)(U + (size_t)batch * CC * HH * WW);
        for (int e = tid; e < CC * MM; e += 256) {
            int c = e / MM, m = e % MM;
            int kh = m / KK, kw = m % KK;
            int h = h0 + kh, w = w0 + kw;
            if (h >= 0 && h < HH && w >= 0 && w < WW) {
                unsigned goff = (unsigned)((((c * HH) + h) * WW + w) * 4);
                unsigned loff = (unsigned)(uintptr_t)(&Xc[c * XS + m]);
                asm volatile("global_load_async_to_lds_b32 %0, %1, %2"
                             :: "v"(loff), "v"(goff), "s"(ubase) : "memory");
            }
        }
        asm volatile("s_wait_asynccnt 0x0" ::: "memory");
    }
    __syncthreads();

    // ---------------- mean, center, per-channel sum of squares ----------------
    if (tid < CC) {
        float s = 0.f;
        #pragma unroll
        for (int m = 0; m < MM; ++m) s += Xc[tid * XS + m];
        float mn = s * (1.f / 25.f);
        mu[tid] = mn;
        uc[tid] = Xc[tid * XS + 12];          // center pixel (M/2)
        float q = 0.f;
        #pragma unroll
        for (int m = 0; m < MM; ++m) {
            float d = Xc[tid * XS + m] - mn;
            Xc[tid * XS + m] = d;
            q += d * d;
        }
        ssq[tid] = q;
    }
    __syncthreads();
    if (tid == 0) {
        float t = 0.f;
        for (int c = 0; c < CC; ++c) t += ssq[c];
        scal[0] = t * (1.f / 25.f);           // trace(R)
    }
    __syncthreads();

    // ---------------- covariance via V_WMMA_F32_16X16X4_F32 ----------------
    // R = Xc (64x32) * Xc^T (32x64) is symmetric: compute only the 10 upper tiles
    // (i<=j), mirror-store off-diagonal tiles. A and B fragments share one loader.
    const float diagAdd  = ALPHA * scal[0] * (1.f / 64.f) + DELTA;
    const float covScale = (1.f - ALPHA) * (1.f / 25.f);
    const int wid   = tid >> 5;
    const int lane  = tid & 31;
    const int lrow  = lane & 15;
    const int khalf = (lane >> 4) << 1;       // lanes 16-31 hold K+2,K+3

    for (int t = wid; t < 10; t += 8) {
        const int code = (int)((0xFBA7653210ULL >> (4 * t)) & 0xF); // (i<<2)|j, i<=j
        const int ti = code >> 2, tj = code & 3;
        v8f acc = {};
        #pragma unroll
        for (int k = 0; k < 8; ++k) {
            const int kb = 4 * k + khalf;
            v2f af, bf;
            af.x = Xc[(16 * ti + lrow) * XS + kb];
            af.y = Xc[(16 * ti + lrow) * XS + kb + 1];
            bf.x = Xc[(16 * tj + lrow) * XS + kb];
            bf.y = Xc[(16 * tj + lrow) * XS + kb + 1];
            acc = __builtin_amdgcn_wmma_f32_16x16x4_f32(
                      false, af, false, bf, (short)0, acc, false, false);
        }
        // C/D layout: lanes0-15 -> rows 0..7 (vgpr r), col=lane; lanes16-31 -> rows 8..15
        const int rbase = 16 * ti + ((lane >> 4) << 3);
        const int col   = 16 * tj + lrow;
        #pragma unroll
        for (int r = 0; r < 8; ++r) {
            int row = rbase + r;
            float val = acc[r] * covScale;
            if (row == col) val += diagAdd;
            Rm[row * RS + col] = val;
            if (ti != tj) Rm[col * RS + row] = val;   // symmetric mirror
        }
    }
    __syncthreads();

    // ---------------- MLP head: a = normalize(softplus(relu(mu@w1+b1)@w2+b2)) ------
    if (tid < HID) {
        float s = b1[tid];
        for (int c = 0; c < CC; ++c) s += mu[c] * w1[c * HID + tid];
        hv[tid] = s > 0.f ? s : 0.f;
    }
    __syncthreads();
    if (tid < CC) {
        float s = b2[tid];
        for (int k = 0; k < HID; ++k) s += hv[k] * w2[k * CC + tid];
        av[tid] = (s > 0.f) ? (s + log1pf(expf(-s))) : log1pf(expf(s));  // softplus
    }
    __syncthreads();
    if (tid == 0) {
        float n2 = 0.f;
        for (int c = 0; c < CC; ++c) n2 += av[c] * av[c];
        scal[1] = 1.f / fmaxf(sqrtf(n2), 1e-6f);
    }
    __syncthreads();
    if (tid < CC) {
        av[tid] *= scal[1];
        dsave[tid] = Rm[tid * RS + tid];      // save R_s diagonal (Cholesky destroys it)
        wt[tid] = av[tid];                    // rhs for forward solve
    }
    __syncthreads();

    // ---------------- in-place Cholesky (lower triangle) ----------------
    for (int kc = 0; kc < CC; ++kc) {
        if (tid == 0) Rm[kc * RS + kc] = sqrtf(Rm[kc * RS + kc]);
        __syncthreads();
        if (tid > kc && tid < CC) Rm[tid * RS + kc] /= Rm[kc * RS + kc];
        __syncthreads();
        int n = CC - 1 - kc;
        for (int idx = tid; idx < n * n; idx += 256) {
            int i = kc + 1 + idx / n;
            int j = kc + 1 + idx % n;
            if (j <= i) Rm[i * RS + j] -= Rm[i * RS + kc] * Rm[j * RS + kc];
        }
        __syncthreads();
    }

    // ---------------- forward solve L z = a ----------------
    for (int kc = 0; kc < CC; ++kc) {
        if (tid == 0) wt[kc] /= Rm[kc * RS + kc];
        __syncthreads();
        if (tid > kc && tid < CC) wt[tid] -= Rm[tid * RS + kc] * wt[kc];
        __syncthreads();
    }
    // ---------------- back solve L^T w~ = z ----------------
    for (int kc = CC - 1; kc >= 0; --kc) {
        if (tid == 0) wt[kc] /= Rm[kc * RS + kc];
        __syncthreads();
        if (tid < kc) wt[tid] -= Rm[kc * RS + tid] * wt[kc];
        __syncthreads();
    }

    // ---------------- MVDR normalize, outputs ----------------
    if (tid == 0) {
        float s = 0.f;
        for (int c = 0; c < CC; ++c) s += av[c] * wt[c];
        scal[2] = 1.f / fmaxf(s, 1e-6f);
    }
    __syncthreads();
    if (tid < CC) wv[tid] = wt[tid] * scal[2];
    __syncthreads();
    if (tid < CC) {   // t = R_s * w  using saved diag + intact upper triangle
        float s = 0.f;
        for (int j = 0; j < CC; ++j) {
            float rij = (j > tid) ? Rm[tid * RS + j]
                                  : ((j == tid) ? dsave[tid] : Rm[j * RS + tid]);
            s += rij * wv[j];
        }
        tv[tid] = s;
    }
    __syncthreads();
    if (tid == 0) {
        float y = 0.f, v = 0.f;
        for (int c = 0; c < CC; ++c) { y += wv[c] * uc[c]; v += wv[c] * tv[c]; }
        y_out[bidx] = y;
        v_out[bidx] = v;
    }
}

// jax.image.resize(method='linear'): src = (dst+0.5)/scale - 0.5, edge-clamped
__global__ __launch_bounds__(256)
void upsample_bilinear_kernel(const float* __restrict__ ysrc,
                              const float* __restrict__ vsrc,
                              float* __restrict__ out)
{
    int idx = blockIdx.x * 256 + threadIdx.x;
    const int per = BB * HH * WW;
    if (idx >= 2 * per) return;
    const float* src = (idx < per) ? ysrc : vsrc;
    int rem = (idx < per) ? idx : idx - per;
    int b = rem / (HH * WW);
    int p = rem % (HH * WW);
    int oy = p / WW, ox = p % WW;
    float sy = oy * 0.5f - 0.25f;
    float sx = ox * 0.5f - 0.25f;
    float fy0 = floorf(sy), fx0 = floorf(sx);
    float wy = sy - fy0, wx = sx - fx0;
    int iy0 = (int)fy0, ix0 = (int)fx0;
    int iy1 = iy0 + 1, ix1 = ix0 + 1;
    iy0 = iy0 < 0 ? 0 : (iy0 > 63 ? 63 : iy0);
    iy1 = iy1 < 0 ? 0 : (iy1 > 63 ? 63 : iy1);
    ix0 = ix0 < 0 ? 0 : (ix0 > 63 ? 63 : ix0);
    ix1 = ix1 < 0 ? 0 : (ix1 > 63 ? 63 : ix1);
    const float* sb = src + b * (HOUT * WOUT);
    float v00 = sb[iy0 * WOUT + ix0], v01 = sb[iy0 * WOUT + ix1];
    float v10 = sb[iy1 * WOUT + ix0], v11 = sb[iy1 * WOUT + ix1];
    float v0 = v00 + (v01 - v00) * wx;
    float v1 = v10 + (v11 - v10) * wx;
    out[idx] = v0 + (v1 - v0) * wy;
}

extern "C" void kernel_launch(void* const* d_in, const int* in_sizes, int n_in,
                              void* d_out, int out_size, void* d_ws, size_t ws_size,
                              hipStream_t stream) {
    const float* U  = (const float*)d_in[0];
    const float* w1 = (const float*)d_in[1];
    const float* b1 = (const float*)d_in[2];
    const float* w2 = (const float*)d_in[3];
    const float* b2 = (const float*)d_in[4];
    float* out = (float*)d_out;

    float* ws_y = (float*)d_ws;               // [B*L]
    float* ws_v = ws_y + (size_t)BB * LL;     // [B*L]

    capon_tile_kernel<<<BB * LL, 256, 0, stream>>>(U, w1, b1, w2, b2, ws_y, ws_v);

    const int total = 2 * BB * HH * WW;
    upsample_bilinear_kernel<<<(total + 255) / 256, 256, 0, stream>>>(ws_y, ws_v, out);
}